// MLPModel_34024730919033
// MI455X (gfx1250) — compile-verified
//
#include <hip/hip_runtime.h>

typedef float v2f __attribute__((ext_vector_type(2)));
typedef float v8f __attribute__((ext_vector_type(8)));

// ---- problem dims ----
constexpr int B  = 1024, S = 12, L = 52, H = 256, CE = 128, CO = 17, O = 2;
constexpr int NROW = B * S;          // 12288 sample rows
constexpr int K1   = L * CE;         // 6656: outer-product z(l,c) region, k = l*128 + c
constexpr int KCE  = K1 + L;         // 6708: obs region end
constexpr int KONE = KCE + CE;       // 6836: ce region end; k==KONE -> constant 1
constexpr int KAUG = 6848;           // padded to 32 * 214
constexpr int NCHUNK  = KAUG / 32;   // 214 chunks of 32 k-rows
constexpr int NCHUNKZ = K1 / 32;     // 208 chunks fully inside the z-region
constexpr int CES = 132;             // ce LDS row stride (b64 bank-conflict-free)
constexpr int OBS = 53;              // obs LDS row stride
constexpr int BPS = 544;             // B LDS pair stride (== 32 mod 64 -> bank split)

// ---------------------------------------------------------------------------
// Build augmented B matrix, PAIR-INTERLEAVED: flat = (k>>1)*512 + h*2 + (k&1)
// so B fragments (rows k, k+1 at one column) are contiguous 8 bytes.
// ---------------------------------------------------------------------------
__global__ void build_baug_kernel(const float* __restrict__ Ww,   // (CE, L*H)
                                  const float* __restrict__ Wb,   // (L*H,)
                                  const float* __restrict__ Bw,   // (CE, H)
                                  const float* __restrict__ Bb,   // (H,)
                                  float* __restrict__ Baug) {
  int idx = blockIdx.x * blockDim.x + threadIdx.x;
  if (idx >= KAUG * H) return;
  int pair = idx >> 9;
  int p    = idx & 1;
  int h    = (idx & 511) >> 1;
  int k    = pair * 2 + p;
  float v;
  if (k < K1) {
    int l = k >> 7, c = k & 127;
    v = Ww[c * (L * H) + l * H + h];
  } else if (k < KCE) {
    v = Wb[(k - K1) * H + h];
  } else if (k < KONE) {
    v = Bw[(k - KCE) * H + h];
  } else if (k == KONE) {
    v = Bb[h];
  } else {
    v = 0.0f;
  }
  Baug[idx] = v;
}

// ---------------------------------------------------------------------------
// Generic fused linear + ReLU: out = relu(in @ W + b), W row-major (K,N)
// ---------------------------------------------------------------------------
__global__ void linear_relu_kernel(const float* __restrict__ in,
                                   const float* __restrict__ W,
                                   const float* __restrict__ bias,
                                   float* __restrict__ out,
                                   int M, int K, int N) {
  int idx = blockIdx.x * blockDim.x + threadIdx.x;
  if (idx >= M * N) return;
  int row = idx / N, j = idx - row * N;
  float acc = bias[j];
  const float* ip = in + row * K;
  for (int k = 0; k < K; ++k) acc = fmaf(ip[k], W[k * N + j], acc);
  out[idx] = fmaxf(acc, 0.0f);
}

// tail-region A element (k >= K1 only): obs rows, ce rows, constant-1, zero pad
__device__ __forceinline__ float tail_a(const float* obs_s, const float* ce_s,
                                        int row, int k) {
  if (k < KCE)  return obs_s[row * OBS + (k - K1)];
  if (k < KONE) return ce_s[row * CES + (k - KCE)];
  return (k == KONE) ? 1.0f : 0.0f;
}

// ---------------------------------------------------------------------------
// Dominant GEMM: emb_pre[12288,256] = A_aug(12288 x 6848) @ Baug(6848 x 256)
// fp32 WMMA 16x16x4.  WG = 256 thr (8 wave32) -> 64 rows x 256 cols per WG,
// wave: 64x32 tile = 8 C fragments.  B staged via double-buffered
// global_load_async_to_lds_b128 (ASYNCcnt), A generated on the fly from the
// rank-1 obs x ce structure held in LDS.
// ---------------------------------------------------------------------------
__global__ void __launch_bounds__(256)
hyper_in_wmma_kernel(const float* __restrict__ obs,     // (NROW, L)
                     const float* __restrict__ ce,      // (NROW, CE)
                     const float* __restrict__ Baug,    // interleaved (KAUG/2, 256, 2)
                     float* __restrict__ emb_pre) {     // (NROW, 256)
  __shared__ float obs_s[64 * OBS];
  __shared__ float ce_s[64 * CES];
  __shared__ float Bs[2][16 * BPS];

  const int tid  = threadIdx.x;
  const int lane = tid & 31;
  const int wave = tid >> 5;       // 0..7
  const int n    = lane & 15;
  const int kh   = lane >> 4;      // 0: K0/K1 half, 1: K2/K3 half (A/B layout)
  const int m    = lane & 15;
  const int rowbase = blockIdx.x * 64;
  const int col0 = wave * 32 + n;
  const int col1 = col0 + 16;

  // ---- stage this WG's 64 rows of obs and ce into LDS ----
  for (int e = tid; e < 64 * L; e += 256) {
    int r = e / L, l = e - r * L;
    obs_s[r * OBS + l] = obs[(rowbase + r) * L + l];
  }
  for (int f = tid; f < 64 * 32; f += 256) {          // 2048 float4s of ce
    int r = f >> 5, c4 = (f & 31) * 4;
    *(float4*)&ce_s[r * CES + c4] = *(const float4*)&ce[(rowbase + r) * CE + c4];
  }

  v8f acc[4][2];
  #pragma unroll
  for (int i = 0; i < 4; ++i) { acc[i][0] = v8f{}; acc[i][1] = v8f{}; }

  // async-stage one 32-k chunk (16 pairs x 512 floats, contiguous in global)
  auto stage = [&](int chunk, int buf) {
    const float* g = Baug + (size_t)chunk * 8192;
    #pragma unroll
    for (int j = 0; j < 8; ++j) {
      int f = tid + j * 256;                 // float4 index within chunk
      int pair = f >> 7;
      int off  = (f & 127) * 4;
      unsigned lds = (unsigned)(uintptr_t)&Bs[buf][pair * BPS + off];
      const float* ga = g + f * 4;
      asm volatile("global_load_async_to_lds_b128 %0, %1, off"
                   :: "v"(lds), "v"(ga) : "memory");
    }
  };

  stage(0, 0);

  for (int chunk = 0; chunk < NCHUNK; ++chunk) {
    asm volatile("s_wait_asynccnt 0x0" ::: "memory");
    __syncthreads();                         // chunk's B ready; prev buf free
    if (chunk + 1 < NCHUNK) stage(chunk + 1, (chunk + 1) & 1);
    const float* Bc = Bs[chunk & 1];

    if (chunk < NCHUNKZ) {
      // pure z-region: l constant within the chunk -> obs factor in registers
      const int l  = chunk >> 2;             // (chunk*32) >> 7
      const int cb0 = (chunk & 3) * 32;      // (chunk*32) & 127
      float o0 = obs_s[(m)      * OBS + l];
      float o1 = obs_s[(m + 16) * OBS + l];
      float o2 = obs_s[(m + 32) * OBS + l];
      float o3 = obs_s[(m + 48) * OBS + l];
      #pragma unroll
      for (int kt = 0; kt < 8; ++kt) {
        const int cb = cb0 + kt * 4 + 2 * kh;
        v2f e0 = *(const v2f*)&ce_s[(m)      * CES + cb];
        v2f e1 = *(const v2f*)&ce_s[(m + 16) * CES + cb];
        v2f e2 = *(const v2f*)&ce_s[(m + 32) * CES + cb];
        v2f e3 = *(const v2f*)&ce_s[(m + 48) * CES + cb];
        const int pr = (kt * 2 + kh) * BPS;
        v2f b0 = *(const v2f*)&Bc[pr + col0 * 2];
        v2f b1 = *(const v2f*)&Bc[pr + col1 * 2];
        v2f a0{o0 * e0.x, o0 * e0.y}, a1{o1 * e1.x, o1 * e1.y};
        v2f a2{o2 * e2.x, o2 * e2.y}, a3{o3 * e3.x, o3 * e3.y};
        acc[0][0] = __builtin_amdgcn_wmma_f32_16x16x4_f32(false, a0, false, b0, (short)0, acc[0][0], false, false);
        acc[0][1] = __builtin_amdgcn_wmma_f32_16x16x4_f32(false, a0, false, b1, (short)0, acc[0][1], false, false);
        acc[1][0] = __builtin_amdgcn_wmma_f32_16x16x4_f32(false, a1, false, b0, (short)0, acc[1][0], false, false);
        acc[1][1] = __builtin_amdgcn_wmma_f32_16x16x4_f32(false, a1, false, b1, (short)0, acc[1][1], false, false);
        acc[2][0] = __builtin_amdgcn_wmma_f32_16x16x4_f32(false, a2, false, b0, (short)0, acc[2][0], false, false);
        acc[2][1] = __builtin_amdgcn_wmma_f32_16x16x4_f32(false, a2, false, b1, (short)0, acc[2][1], false, false);
        acc[3][0] = __builtin_amdgcn_wmma_f32_16x16x4_f32(false, a3, false, b0, (short)0, acc[3][0], false, false);
        acc[3][1] = __builtin_amdgcn_wmma_f32_16x16x4_f32(false, a3, false, b1, (short)0, acc[3][1], false, false);
      }
    } else {
      // tail: obs / ce / 1 / zero rows
      const int k0 = chunk * 32;
      #pragma unroll
      for (int kt = 0; kt < 8; ++kt) {
        const int kg = k0 + kt * 4 + 2 * kh;
        v2f a0{tail_a(obs_s, ce_s, m,      kg), tail_a(obs_s, ce_s, m,      kg + 1)};
        v2f a1{tail_a(obs_s, ce_s, m + 16, kg), tail_a(obs_s, ce_s, m + 16, kg + 1)};
        v2f a2{tail_a(obs_s, ce_s, m + 32, kg), tail_a(obs_s, ce_s, m + 32, kg + 1)};
        v2f a3{tail_a(obs_s, ce_s, m + 48, kg), tail_a(obs_s, ce_s, m + 48, kg + 1)};
        const int pr = (kt * 2 + kh) * BPS;
        v2f b0 = *(const v2f*)&Bc[pr + col0 * 2];
        v2f b1 = *(const v2f*)&Bc[pr + col1 * 2];
        acc[0][0] = __builtin_amdgcn_wmma_f32_16x16x4_f32(false, a0, false, b0, (short)0, acc[0][0], false, false);
        acc[0][1] = __builtin_amdgcn_wmma_f32_16x16x4_f32(false, a0, false, b1, (short)0, acc[0][1], false, false);
        acc[1][0] = __builtin_amdgcn_wmma_f32_16x16x4_f32(false, a1, false, b0, (short)0, acc[1][0], false, false);
        acc[1][1] = __builtin_amdgcn_wmma_f32_16x16x4_f32(false, a1, false, b1, (short)0, acc[1][1], false, false);
        acc[2][0] = __builtin_amdgcn_wmma_f32_16x16x4_f32(false, a2, false, b0, (short)0, acc[2][0], false, false);
        acc[2][1] = __builtin_amdgcn_wmma_f32_16x16x4_f32(false, a2, false, b1, (short)0, acc[2][1], false, false);
        acc[3][0] = __builtin_amdgcn_wmma_f32_16x16x4_f32(false, a3, false, b0, (short)0, acc[3][0], false, false);
        acc[3][1] = __builtin_amdgcn_wmma_f32_16x16x4_f32(false, a3, false, b1, (short)0, acc[3][1], false, false);
      }
    }
  }

  // C layout: VGPR v -> M = v (lanes 0-15) / v+8 (lanes 16-31), N = lane&15
  #pragma unroll
  for (int s = 0; s < 4; ++s) {
    const float* p0 = (const float*)&acc[s][0];
    const float* p1 = (const float*)&acc[s][1];
    #pragma unroll
    for (int v = 0; v < 8; ++v) {
      int r = rowbase + s * 16 + v + 8 * kh;
      emb_pre[r * 256 + col0] = p0[v];
      emb_pre[r * 256 + col1] = p1[v];
    }
  }
}

// ---------------------------------------------------------------------------
// mask -> relu -> mean over active limbs: emb[b,h]
// ---------------------------------------------------------------------------
__global__ void aggregate_kernel(const float* __restrict__ emb_pre,
                                 const int* __restrict__ mask,
                                 float* __restrict__ emb) {
  int idx = blockIdx.x * blockDim.x + threadIdx.x;
  if (idx >= B * H) return;
  int b = idx >> 8, h = idx & 255;
  float acc = 0.0f; int cnt = 0;
  for (int s = 0; s < S; ++s) {
    if (mask[b * S + s] == 0) {
      acc += fmaxf(emb_pre[(b * S + s) * H + h], 0.0f);
      cnt++;
    }
  }
  emb[idx] = acc / (float)cnt;
}

// ---------------------------------------------------------------------------
// Output hypernet, algebraically collapsed:
//   q[b,c,o] = sum_h emb2[b,h] * hn_out_w_w[c, h*O+o]
//   r[b,o]   = sum_h emb2[b,h] * hn_out_w_b[h*O+o] + hn_out_b_b[o]
//   out[b,s,o] = sum_c ceo[b,s,c] * (q[b,c,o] + hn_out_b_w[c,o]) + r[b,o]
// ---------------------------------------------------------------------------
__global__ void qproj_kernel(const float* __restrict__ emb2,
                             const float* __restrict__ Ww,
                             float* __restrict__ q) {
  int idx = blockIdx.x * blockDim.x + threadIdx.x;
  if (idx >= B * CE * O) return;
  int b = idx >> 8;            // CE*O == 256
  int co = idx & 255;
  int c = co >> 1, o = co & 1;
  const float* e = emb2 + b * H;
  float acc = 0.0f;
  for (int h = 0; h < H; ++h) acc = fmaf(e[h], Ww[c * (H * O) + h * O + o], acc);
  q[idx] = acc;
}

__global__ void rproj_kernel(const float* __restrict__ emb2,
                             const float* __restrict__ Wb,
                             const float* __restrict__ Bb,
                             float* __restrict__ r) {
  int idx = blockIdx.x * blockDim.x + threadIdx.x;
  if (idx >= B * O) return;
  int b = idx >> 1, o = idx & 1;
  const float* e = emb2 + b * H;
  float acc = Bb[o];
  for (int h = 0; h < H; ++h) acc = fmaf(e[h], Wb[h * O + o], acc);
  r[idx] = acc;
}

__global__ void final_kernel(const float* __restrict__ ceo,
                             const float* __restrict__ q,
                             const float* __restrict__ r,
                             const float* __restrict__ Bw,
                             float* __restrict__ out) {
  int idx = blockIdx.x * blockDim.x + threadIdx.x;
  if (idx >= NROW * O) return;
  int row = idx >> 1, o = idx & 1;
  int b = row / S;
  const float* cp = ceo + row * CE;
  const float* qp = q + b * (CE * O);
  float acc = r[b * O + o];
  for (int c = 0; c < CE; ++c)
    acc = fmaf(cp[c], qp[c * 2 + o] + Bw[c * 2 + o], acc);
  out[idx] = acc;
}

// ---------------------------------------------------------------------------
extern "C" void kernel_launch(void* const* d_in, const int* in_sizes, int n_in,
                              void* d_out, int out_size, void* d_ws, size_t ws_size,
                              hipStream_t stream) {
  (void)in_sizes; (void)n_in; (void)out_size; (void)ws_size;
  const float* obs        = (const float*)d_in[0];
  const int*   obs_mask   = (const int*)  d_in[1];
  const float* obs_ctx    = (const float*)d_in[2];
  const float* ce_in_w1   = (const float*)d_in[3];
  const float* ce_in_b1   = (const float*)d_in[4];
  const float* ce_in_w2   = (const float*)d_in[5];
  const float* ce_in_b2   = (const float*)d_in[6];
  const float* hn_in_w_w  = (const float*)d_in[7];
  const float* hn_in_w_b  = (const float*)d_in[8];
  const float* hn_in_b_w  = (const float*)d_in[9];
  const float* hn_in_b_b  = (const float*)d_in[10];
  const float* hid_w1     = (const float*)d_in[11];
  const float* hid_b1     = (const float*)d_in[12];
  const float* hid_w2     = (const float*)d_in[13];
  const float* hid_b2     = (const float*)d_in[14];
  const float* ce_out_w1  = (const float*)d_in[15];
  const float* ce_out_b1  = (const float*)d_in[16];
  const float* ce_out_w2  = (const float*)d_in[17];
  const float* ce_out_b2  = (const float*)d_in[18];
  const float* hn_out_w_w = (const float*)d_in[19];
  const float* hn_out_w_b = (const float*)d_in[20];
  const float* hn_out_b_w = (const float*)d_in[21];
  const float* hn_out_b_b = (const float*)d_in[22];
  float* out = (float*)d_out;

  // workspace layout (floats), ~42.7 MB total
  float* ws      = (float*)d_ws;
  float* h1      = ws;                      // NROW*CE (temp, reused)
  float* ce      = h1      + NROW * CE;     // NROW*CE
  float* ceo     = ce      + NROW * CE;     // NROW*CE
  float* Baug    = ceo     + NROW * CE;     // KAUG*H (pair-interleaved)
  float* emb_pre = Baug    + KAUG * H;      // NROW*H
  float* emb     = emb_pre + NROW * H;      // B*H
  float* embh    = emb     + B * H;         // B*H
  float* emb2    = embh    + B * H;         // B*H
  float* qbuf    = emb2    + B * H;         // B*CE*O
  float* rbuf    = qbuf    + B * CE * O;    // B*O

  const int T = 256;

  // augmented, pair-interleaved B matrix for the hypernet-input GEMM
  build_baug_kernel<<<(KAUG * H) / T, T, 0, stream>>>(hn_in_w_w, hn_in_w_b,
                                                      hn_in_b_w, hn_in_b_b, Baug);

  // input context encoder: (NROW,17) -> (NROW,128) -> (NROW,128)
  linear_relu_kernel<<<(NROW * CE) / T, T, 0, stream>>>(obs_ctx, ce_in_w1, ce_in_b1, h1, NROW, CO, CE);
  linear_relu_kernel<<<(NROW * CE) / T, T, 0, stream>>>(h1, ce_in_w2, ce_in_b2, ce, NROW, CE, CE);

  // dominant 42-GFLOP GEMM on the fp32 WMMA path (async double-buffered B)
  hyper_in_wmma_kernel<<<NROW / 64, T, 0, stream>>>(obs, ce, Baug, emb_pre);

  // output context encoder (independent; reuses h1)
  linear_relu_kernel<<<(NROW * CE) / T, T, 0, stream>>>(obs_ctx, ce_out_w1, ce_out_b1, h1, NROW, CO, CE);
  linear_relu_kernel<<<(NROW * CE) / T, T, 0, stream>>>(h1, ce_out_w2, ce_out_b2, ceo, NROW, CE, CE);

  // mask -> relu -> mean over limbs
  aggregate_kernel<<<(B * H) / T, T, 0, stream>>>(emb_pre, obs_mask, emb);

  // hidden MLP (1024x256 @ 256x256, twice)
  linear_relu_kernel<<<(B * H) / T, T, 0, stream>>>(emb, hid_w1, hid_b1, embh, B, H, H);
  linear_relu_kernel<<<(B * H) / T, T, 0, stream>>>(embh, hid_w2, hid_b2, emb2, B, H, H);

  // collapsed output hypernet
  qproj_kernel<<<(B * CE * O) / T, T, 0, stream>>>(emb2, hn_out_w_w, qbuf);
  rproj_kernel<<<(B * O) / T, T, 0, stream>>>(emb2, hn_out_w_b, hn_out_b_b, rbuf);
  final_kernel<<<(NROW * O) / T, T, 0, stream>>>(ceo, qbuf, rbuf, hn_out_b_w, out);
}